// PoseMixtureVAE_74423193305293
// MI455X (gfx1250) — compile-verified
//
#include <hip/hip_runtime.h>
#include <hip/hip_bf16.h>

// ---------------------------------------------------------------------------
// PoseMixtureVAE forward, fp32, CDNA5 (gfx1250).
//
// All heavy math runs on V_WMMA_F32_16X16X4_F32 (fp32 matrix cores).
// ~6.6 GFLOP, weights ~7 MB (L2-resident on 192 MB L2), ~25 MB activation
// traffic => bandwidth bound; fp32 WMMA preserves fp32 parity.
//
// GEMMs = sums of BRANCH-FREE segment GEMMs:
//   [A1|A2] @ W        = A1 @ W[0:k1] + A2 @ W[k1:k1+k2]
//   MoE (coeff-scaled) = sum_e coeff[b,e] * ([z|prev] @ W_e)
// N is a TEMPLATE parameter and all loads use 32-bit element indices off
// uniform (SGPR) base pointers => GLOBAL (not FLAT) loads in GVS form with
// immediate offsets; index bumps are 32-bit adds. K tails use one guarded
// step; N edges clamp loads and guard stores only. Each wave computes a
// 16x(16*NT) output tile.
// ---------------------------------------------------------------------------

typedef __attribute__((ext_vector_type(2))) float v2f;
typedef __attribute__((ext_vector_type(8))) float v8f;

#define BATCH 2048
#define FRAME 267
#define LAT   32
#define HID   256
#define NE    6
#define GH    64

__device__ __forceinline__ float elu1(float x) {
  return x > 0.f ? x : expm1f(x);
}

__device__ __forceinline__ v8f wmma4(v2f a, v2f b, v8f c) {
  return __builtin_amdgcn_wmma_f32_16x16x4_f32(
      false, a, false, b, (short)0, c, false, false);
}

// acc[j] += (scale * Aseg[16 rows, len]) @ Wseg[len, N] (cols pre-clamped).
// All accesses are 32-bit indices off Aseg/Wseg (global, GVS addressing).
template <int N, bool MOE, int NT>
__device__ __forceinline__ void seg_gemm(
    const float* __restrict__ Aseg, int aBase,  // rowA*lda
    int len, float scale,
    const float* __restrict__ Wseg,             // uniform segment base
    const int* col, int h, v8f* acc)
{
  int wIdx[NT];
  #pragma unroll
  for (int j = 0; j < NT; ++j) wIdx[j] = 2 * h * N + col[j];
  const int a0 = aBase + 2 * h;

  const int len4 = len & ~3;
  int k = 0;
  #pragma unroll 8
  for (; k < len4; k += 4) {
    v2f a;
    a.x = Aseg[a0 + k];
    a.y = Aseg[a0 + k + 1];
    if (MOE) { a.x *= scale; a.y *= scale; }
    #pragma unroll
    for (int j = 0; j < NT; ++j) {
      v2f b;
      b.x = Wseg[wIdx[j]];
      b.y = Wseg[wIdx[j] + N];
      acc[j] = wmma4(a, b, acc[j]);
      wIdx[j] += 4 * N;
    }
  }
  if (k < len) {                       // single guarded tail (len % 4 != 0)
    const int kk = k + 2 * h;
    v2f a;
    a.x = (kk     < len) ? Aseg[aBase + kk]     : 0.f;
    a.y = (kk + 1 < len) ? Aseg[aBase + kk + 1] : 0.f;
    if (MOE) { a.x *= scale; a.y *= scale; }
    const int r0 = ((kk     < len) ? kk     : len - 1) - kk;  // <= 0
    const int r1 = ((kk + 1 < len) ? kk + 1 : len - 1) - kk;  // 1, 0, or -1
    #pragma unroll
    for (int j = 0; j < NT; ++j) {
      v2f b;                            // clamped rows: 0 * finite = 0
      b.x = Wseg[wIdx[j] + r0 * N];
      b.y = Wseg[wIdx[j] + r1 * N];
      acc[j] = wmma4(a, b, acc[j]);
    }
  }
}

// C[B,N] = act( sum_e coeff[b,e] * ([A1|A2] @ W_e) + bias [+ bias2d] )
template <int N, bool MOE, int NT>
__global__ void __launch_bounds__(32)
gemm_wmma_f32(const float* __restrict__ A1, int k1,
              const float* __restrict__ A2, int k2,
              const float* __restrict__ coeff,   // [B,NE] (MOE only)
              const float* __restrict__ W,       // [(E*)K, N] row-major
              const float* __restrict__ bias,    // [N] or null
              const float* __restrict__ bias2d,  // [B,N] or null
              float* __restrict__ C,             // [B,N]
              int act)
{
  const int lane = threadIdx.x;
  const int h    = lane >> 4;
  const int ln   = lane & 15;
  const int m0   = blockIdx.y * 16;
  const int n0   = blockIdx.x * (16 * NT);
  const int rowA = m0 + ln;

  int col[NT];
  #pragma unroll
  for (int j = 0; j < NT; ++j) {
    const int cb = n0 + 16 * j + ln;
    col[j] = (cb < N) ? cb : N - 1;    // clamp for loads
  }

  v8f acc[NT];
  #pragma unroll
  for (int j = 0; j < NT; ++j) acc[j] = (v8f){0.f,0.f,0.f,0.f,0.f,0.f,0.f,0.f};

  const int a1Base = rowA * k1;
  const int a2Base = rowA * k2;
  const int I = k1 + k2;
  const int nExp = MOE ? NE : 1;

  for (int e = 0; e < nExp; ++e) {
    const float scale = MOE ? coeff[rowA * NE + e] : 1.f;
    const float* Wb = W + (size_t)e * I * N;
    seg_gemm<N, MOE, NT>(A1, a1Base, k1, scale, Wb, col, h, acc);
    if (k2)
      seg_gemm<N, MOE, NT>(A2, a2Base, k2, scale, Wb + (size_t)k1 * N, col, h,
                           acc);
  }

  // Epilogue: NT 16-col subtiles, store-guarded at the N edge.
  #pragma unroll
  for (int j = 0; j < NT; ++j) {
    const int colB = n0 + 16 * j + ln;
    if (colB >= N) continue;
    #pragma unroll
    for (int i = 0; i < 8; ++i) {
      const int row = m0 + i + 8 * h;
      float y = acc[j][i];
      if (bias)   y += bias[colB];
      if (bias2d) y += bias2d[row * N + colB];
      if (act)    y = elu1(y);
      C[row * N + colB] = y;
    }
  }
}

// z = m + eps * exp(0.5*v)
__global__ void reparam_kernel(const float* __restrict__ m,
                               const float* __restrict__ v,
                               const float* __restrict__ eps,
                               float* __restrict__ z, int n) {
  int i = blockIdx.x * blockDim.x + threadIdx.x;
  if (i < n) z[i] = m[i] + eps[i] * expf(0.5f * v[i]);
}

// coeff = softmax(g @ g3_w + g3_b), N=6 — one thread per row.
__global__ void gate3_softmax_kernel(const float* __restrict__ g,
                                     const float* __restrict__ w,   // [GH,NE]
                                     const float* __restrict__ b,   // [NE]
                                     float* __restrict__ coeff) {
  int row = blockIdx.x * blockDim.x + threadIdx.x;
  if (row >= BATCH) return;
  float lg[NE];
  #pragma unroll
  for (int e = 0; e < NE; ++e) lg[e] = b[e];
  for (int i = 0; i < GH; ++i) {
    float gv = g[row * GH + i];
    #pragma unroll
    for (int e = 0; e < NE; ++e) lg[e] = fmaf(gv, w[i * NE + e], lg[e]);
  }
  float mx = lg[0];
  #pragma unroll
  for (int e = 1; e < NE; ++e) mx = fmaxf(mx, lg[e]);
  float s = 0.f, ex[NE];
  #pragma unroll
  for (int e = 0; e < NE; ++e) { ex[e] = expf(lg[e] - mx); s += ex[e]; }
  float inv = 1.f / s;
  #pragma unroll
  for (int e = 0; e < NE; ++e) coeff[row * NE + e] = ex[e] * inv;
}

// mb{0,1,2}[b,:] = coeff[b,:] @ b{0,1,2}  (per-row mixed decoder biases)
__global__ void mixed_bias_kernel(const float* __restrict__ coeff,
                                  const float* __restrict__ b0,   // [NE,HID]
                                  const float* __restrict__ b1,   // [NE,HID]
                                  const float* __restrict__ b2,   // [NE,FRAME]
                                  float* __restrict__ mb0,
                                  float* __restrict__ mb1,
                                  float* __restrict__ mb2) {
  int row = blockIdx.x;
  float cf[NE];
  #pragma unroll
  for (int e = 0; e < NE; ++e) cf[e] = coeff[row * NE + e];
  for (int j = threadIdx.x; j < HID; j += blockDim.x) {
    float s0 = 0.f, s1 = 0.f;
    #pragma unroll
    for (int e = 0; e < NE; ++e) {
      s0 = fmaf(cf[e], b0[e * HID + j], s0);
      s1 = fmaf(cf[e], b1[e * HID + j], s1);
    }
    mb0[row * HID + j] = s0;
    mb1[row * HID + j] = s1;
  }
  for (int j = threadIdx.x; j < FRAME; j += blockDim.x) {
    float s2 = 0.f;
    #pragma unroll
    for (int e = 0; e < NE; ++e) s2 = fmaf(cf[e], b2[e * FRAME + j], s2);
    mb2[row * FRAME + j] = s2;
  }
}

template <int N, int NT>
static inline dim3 gemm_grid() {
  return dim3((N + 16 * NT - 1) / (16 * NT), BATCH / 16);
}

extern "C" void kernel_launch(void* const* d_in, const int* in_sizes, int n_in,
                              void* d_out, int out_size, void* d_ws, size_t ws_size,
                              hipStream_t stream) {
  const float* x      = (const float*)d_in[0];
  const float* c      = (const float*)d_in[1];
  const float* eps    = (const float*)d_in[2];
  const float* efc1_w = (const float*)d_in[3];
  const float* efc1_b = (const float*)d_in[4];
  const float* efc2_w = (const float*)d_in[5];
  const float* efc2_b = (const float*)d_in[6];
  const float* em_w   = (const float*)d_in[7];
  const float* em_b   = (const float*)d_in[8];
  const float* ev_w   = (const float*)d_in[9];
  const float* ev_b   = (const float*)d_in[10];
  const float* g1_w   = (const float*)d_in[11];
  const float* g1_b   = (const float*)d_in[12];
  const float* g2_w   = (const float*)d_in[13];
  const float* g2_b   = (const float*)d_in[14];
  const float* g3_w   = (const float*)d_in[15];
  const float* g3_b   = (const float*)d_in[16];
  const float* w0     = (const float*)d_in[17];  // [NE,299,HID] flat
  const float* b0     = (const float*)d_in[18];
  const float* w1     = (const float*)d_in[19];  // [NE,288,HID] flat
  const float* b1     = (const float*)d_in[20];
  const float* w2     = (const float*)d_in[21];  // [NE,288,FRAME] flat
  const float* b2     = (const float*)d_in[22];

  float* out   = (float*)d_out;                 // [B,FRAME]
  float* out_m = out + BATCH * FRAME;           // [B,LAT]
  float* out_v = out_m + BATCH * LAT;           // [B,LAT]

  // Workspace carve (~12 MB; h1/h2 reused for decoder intermediates)
  float* ws    = (float*)d_ws;
  float* h1    = ws;                  // [B,HID]  (reused as dec layer0 out)
  float* h2    = h1  + BATCH * HID;   // [B,HID]  (reused as dec layer1 out)
  float* z     = h2  + BATCH * HID;   // [B,LAT]
  float* g1o   = z   + BATCH * LAT;   // [B,GH]
  float* g2o   = g1o + BATCH * GH;    // [B,GH]
  float* coeff = g2o + BATCH * GH;    // [B,NE]
  float* mb0   = coeff + BATCH * NE;  // [B,HID]
  float* mb1   = mb0 + BATCH * HID;   // [B,HID]
  float* mb2   = mb1 + BATCH * HID;   // [B,FRAME]

  // --- Encoder ---
  gemm_wmma_f32<HID, false, 4><<<gemm_grid<HID, 4>(), 32, 0, stream>>>(
      x, FRAME, c, FRAME, nullptr, efc1_w, efc1_b, nullptr, h1, 1);
  gemm_wmma_f32<HID, false, 4><<<gemm_grid<HID, 4>(), 32, 0, stream>>>(
      x, FRAME, h1, HID, nullptr, efc2_w, efc2_b, nullptr, h2, 1);
  gemm_wmma_f32<LAT, false, 2><<<gemm_grid<LAT, 2>(), 32, 0, stream>>>(
      x, FRAME, h2, HID, nullptr, em_w, em_b, nullptr, out_m, 0);
  gemm_wmma_f32<LAT, false, 2><<<gemm_grid<LAT, 2>(), 32, 0, stream>>>(
      x, FRAME, h2, HID, nullptr, ev_w, ev_b, nullptr, out_v, 0);
  reparam_kernel<<<(BATCH * LAT + 255) / 256, 256, 0, stream>>>(
      out_m, out_v, eps, z, BATCH * LAT);

  // --- Gating network ---
  gemm_wmma_f32<GH, false, 2><<<gemm_grid<GH, 2>(), 32, 0, stream>>>(
      z, LAT, c, FRAME, nullptr, g1_w, g1_b, nullptr, g1o, 1);
  gemm_wmma_f32<GH, false, 2><<<gemm_grid<GH, 2>(), 32, 0, stream>>>(
      g1o, GH, nullptr, 0, nullptr, g2_w, g2_b, nullptr, g2o, 1);
  gate3_softmax_kernel<<<(BATCH + 255) / 256, 256, 0, stream>>>(
      g2o, g3_w, g3_b, coeff);
  mixed_bias_kernel<<<BATCH, 256, 0, stream>>>(coeff, b0, b1, b2, mb0, mb1, mb2);

  // --- Mixed decoder: one coeff-scaled GEMM per layer ---
  gemm_wmma_f32<HID, true, 4><<<gemm_grid<HID, 4>(), 32, 0, stream>>>(
      z, LAT, c, FRAME, coeff, w0, nullptr, mb0, h1, 1);
  gemm_wmma_f32<HID, true, 4><<<gemm_grid<HID, 4>(), 32, 0, stream>>>(
      z, LAT, h1, HID, coeff, w1, nullptr, mb1, h2, 1);
  gemm_wmma_f32<FRAME, true, 4><<<gemm_grid<FRAME, 4>(), 32, 0, stream>>>(
      z, LAT, h2, HID, coeff, w2, nullptr, mb2, out, 0);
}